// MultiRetrievalAugmentedEmbeddingV4_74783970558240
// MI455X (gfx1250) — compile-verified
//
#include <hip/hip_runtime.h>
#include <hip/hip_bf16.h>
#include <math.h>

// ---------------------------------------------------------------------------
// MultiRetrievalAugmentedEmbeddingV4 for MI455X (gfx1250, wave32)
//
// Work: score = clip(cos(q, r_n), 0, 1) over N=50000 library rows (768-d),
// softmax over N, top-25 mask, weighted sum of n_answ rows (384-d), dot with
// 3 option vectors. aud/ocr branches are gated by exactly 0.0 -> skipped.
//
// Phase 1 (bandwidth-bound, 153.6 MB single pass): each wave owns a 16-row
// tile and accumulates dot(q, r_n) for 16 rows via V_WMMA_F32_16X16X4_F32
// (fp32 matrix pipe, no precision loss), while accumulating per-row sum of
// squares in VALU from the same loaded data.
// Phase 2: one 1024-thread workgroup, scores staged in LDS (200 KB < 320 KB
// WGP LDS), softmax stats + 25 iterative argmax + final tiny GEMV.
// ---------------------------------------------------------------------------

typedef __attribute__((ext_vector_type(2))) float v2f;
typedef __attribute__((ext_vector_type(8))) float v8f;

#define NLIB   50000
#define QD     768
#define OD     384
#define TOPK   25
#define TILES  (NLIB / 16)           // 3125 exactly
#define WPB    8                     // waves per block (256 threads)
#define P1_BLOCKS ((TILES + WPB - 1) / WPB)

// ---------------------------------------------------------------------------
// Kernel 0: normalize query, compute vid gate.
// ---------------------------------------------------------------------------
__global__ __launch_bounds__(256, 1)
void prep_kernel(const float* __restrict__ v, const float* __restrict__ temp_vid,
                 float* __restrict__ qn, float* __restrict__ gate)
{
    __shared__ float red[256];
    __shared__ float inv_s;
    const int tid = threadIdx.x;
    float s = 0.f;
    for (int i = tid; i < QD; i += 256) { float x = v[i]; s = fmaf(x, x, s); }
    red[tid] = s;
    __syncthreads();
    for (int off = 128; off > 0; off >>= 1) {
        if (tid < off) red[tid] += red[tid + off];
        __syncthreads();
    }
    if (tid == 0) {
        inv_s   = 1.0f / fmaxf(sqrtf(red[0]), 1e-12f);
        gate[0] = 2.0f / (1.0f + expf(-temp_vid[0]));   // sigmoid * 2
    }
    __syncthreads();
    const float inv = inv_s;
    for (int i = tid; i < QD; i += 256) qn[i] = v[i] * inv;
}

// ---------------------------------------------------------------------------
// Kernel 1: streaming WMMA scoring pass.
// Wave handles one 16-row tile. A = replicated normalized query chunk
// (16x4 fp32 layout: lanes 0-15 -> K=0,1 ; lanes 16-31 -> K=2,3),
// B = 16 library rows' K-chunk (N = lane&15). D[m][n] identical across m,
// so c[0] in lane l = dot(q, row (lane&15)). sumsq accumulated in VALU,
// halves combined with shfl_xor(16).
// ---------------------------------------------------------------------------
__global__ __launch_bounds__(256, 1)
void score_wmma_kernel(const float* __restrict__ n_feats,
                       const float* __restrict__ qn,
                       float* __restrict__ scores)
{
    __shared__ float qs[QD];
    for (int i = threadIdx.x; i < QD; i += 256) qs[i] = qn[i];
    __syncthreads();

    const int wave = threadIdx.x >> 5;
    const int lane = threadIdx.x & 31;
    const int tile = blockIdx.x * WPB + wave;
    if (tile >= TILES) return;                 // wave-uniform, EXEC stays full

    const int row  = tile * 16 + (lane & 15);
    const int koff = (lane >> 4) << 1;         // 0 for lanes 0-15, 2 for 16-31
    const float* __restrict__ rp = n_feats + (size_t)row * QD + koff;
    const float* __restrict__ qp = qs + koff;

    v8f   c   = {0.f, 0.f, 0.f, 0.f, 0.f, 0.f, 0.f, 0.f};
    float ssq = 0.f;

    #pragma unroll 8
    for (int k = 0; k < QD; k += 4) {
        v2f b; b.x = rp[k]; b.y = rp[k + 1];   // global b64 per lane
        v2f a; a.x = qp[k]; a.y = qp[k + 1];   // LDS b64 per lane
        // D = A * B + C, fp32 matrix pipe (16x16x4)
        c = __builtin_amdgcn_wmma_f32_16x16x4_f32(
                /*neg_a=*/false, a, /*neg_b=*/false, b,
                /*c_mod=*/(short)0, c, /*reuse_a=*/false, /*reuse_b=*/false);
        ssq = fmaf(b.x, b.x, fmaf(b.y, b.y, ssq));
    }

    const float ss  = ssq + __shfl_xor(ssq, 16, 32);   // combine lane halves (K=0,1 / K=2,3)
    const float dot = c[0];                            // row (lane&15) dot product
    float s = dot / fmaxf(sqrtf(ss), 1e-12f);          // q already normalized
    s = fminf(fmaxf(s, 0.f), 1.f);                     // clip to [0,1]
    if (lane < 16) scores[row] = s;
}

// ---------------------------------------------------------------------------
// Kernel 2: softmax stats + iterative top-25 + weighted sum + option dots.
// Single 1024-thread workgroup; scores live in LDS (200 KB).
// Dynamic LDS layout (floats): [0,N) scores | [N,N+1024) redv |
// [N+1024,N+2048) redi (as int) | [N+2048,N+2048+384) oia
// ---------------------------------------------------------------------------
__global__ __launch_bounds__(1024, 1)
void topk_gather_kernel(const float* __restrict__ scores_g,
                        const float* __restrict__ gate_p,
                        const float* __restrict__ n_answ,
                        const float* __restrict__ o,
                        float* __restrict__ out)
{
    extern __shared__ float sm[];
    float* sc   = sm;
    float* redv = sm + NLIB;
    int*   redi = (int*)(sm + NLIB + 1024);
    float* oia  = sm + NLIB + 2048;
    const int tid = threadIdx.x;

    for (int i = tid; i < NLIB; i += 1024) sc[i] = scores_g[i];
    __syncthreads();

    // ---- global max ----
    float m = -1e30f;
    for (int i = tid; i < NLIB; i += 1024) m = fmaxf(m, sc[i]);
    redv[tid] = m;
    __syncthreads();
    for (int off = 512; off > 0; off >>= 1) {
        if (tid < off) redv[tid] = fmaxf(redv[tid], redv[tid + off]);
        __syncthreads();
    }
    const float gmax = redv[0];
    __syncthreads();

    // ---- sum of exp ----
    float s = 0.f;
    for (int i = tid; i < NLIB; i += 1024) s += expf(sc[i] - gmax);
    redv[tid] = s;
    __syncthreads();
    for (int off = 512; off > 0; off >>= 1) {
        if (tid < off) redv[tid] += redv[tid + off];
        __syncthreads();
    }
    const float gsum = redv[0];
    __syncthreads();

    const float gate = gate_p[0];
    float acc = 0.f;   // this thread's oia dim (tid < 384)

    // ---- 25 iterative argmax passes (tie -> lower index, matching lax.top_k) ----
    for (int t = 0; t < TOPK; t++) {
        float bm = -1e30f;
        int   bi = 0x7fffffff;
        for (int i = tid; i < NLIB; i += 1024) {
            float val = sc[i];
            if (val > bm || (val == bm && i < bi)) { bm = val; bi = i; }
        }
        redv[tid] = bm; redi[tid] = bi;
        __syncthreads();
        for (int off = 512; off > 0; off >>= 1) {
            if (tid < off) {
                float v2 = redv[tid + off]; int i2 = redi[tid + off];
                if (v2 > redv[tid] || (v2 == redv[tid] && i2 < redi[tid])) {
                    redv[tid] = v2; redi[tid] = i2;
                }
            }
            __syncthreads();
        }
        const int   j  = redi[0];
        const float sv = redv[0];
        __syncthreads();

        const float w = expf(sv - gmax) / gsum * gate;
        if (tid < OD) acc = fmaf(w, n_answ[(size_t)j * OD + tid], acc);
        if (tid == 0) sc[j] = -1e30f;    // exclude from next pass
        __syncthreads();
    }

    if (tid < OD) oia[tid] = acc;
    __syncthreads();

    // ---- scores[k] = dot(oia, o[k]) for k = 0..2 ----
    for (int k = 0; k < 3; k++) {
        redv[tid] = (tid < OD) ? oia[tid] * o[k * OD + tid] : 0.f;
        __syncthreads();
        for (int off = 512; off > 0; off >>= 1) {
            if (tid < off) redv[tid] += redv[tid + off];
            __syncthreads();
        }
        if (tid == 0) out[k] = redv[0];
        __syncthreads();
    }
}

// ---------------------------------------------------------------------------
// Launch: inputs in setup_inputs() order:
// 0:v 1:n_feats 2:aud 3:n_auds 4:ocr 5:n_ocrs 6:o 7:n_answ
// 8:temp_vid 9:temp_aud 10:temp_ocr    (aud/ocr gated by exact 0 -> unused)
// ---------------------------------------------------------------------------
extern "C" void kernel_launch(void* const* d_in, const int* in_sizes, int n_in,
                              void* d_out, int out_size, void* d_ws, size_t ws_size,
                              hipStream_t stream)
{
    const float* v        = (const float*)d_in[0];
    const float* n_feats  = (const float*)d_in[1];
    const float* o        = (const float*)d_in[6];
    const float* n_answ   = (const float*)d_in[7];
    const float* temp_vid = (const float*)d_in[8];
    float* out = (float*)d_out;

    float* wsf    = (float*)d_ws;
    float* qn     = wsf;          // 768 floats
    float* gate   = wsf + QD;     // 1 float (padded to 1024)
    float* scores = wsf + 1024;   // 50000 floats

    prep_kernel<<<1, 256, 0, stream>>>(v, temp_vid, qn, gate);
    score_wmma_kernel<<<P1_BLOCKS, 256, 0, stream>>>(n_feats, qn, scores);

    const size_t smem_bytes = (size_t)(NLIB + 2048 + OD) * sizeof(float); // ~205 KB
    (void)hipFuncSetAttribute((const void*)topk_gather_kernel,
                              hipFuncAttributeMaxDynamicSharedMemorySize,
                              (int)smem_bytes);
    topk_gather_kernel<<<1, 1024, smem_bytes, stream>>>(scores, gate, n_answ, o, out);
}